// Network_57037165691650
// MI455X (gfx1250) — compile-verified
//
#include <hip/hip_runtime.h>
#include <hip/hip_bf16.h>
#include <stdint.h>

#define CLASS_NUM 80
#define TOPK_N    1000
// Levels needing top-k (level 4 with 576 rows is kept whole)
static const int LVL_N[4] = {147456, 36864, 9216, 2304};
static const int KOFF[4]  = {0, 147456, 184320, 193536};  // key offsets (uints)
#define KEYS_TOTAL 195840
#define N_KEPT     4576                  // 4*1000 + 576

typedef int v4i_vs __attribute__((vector_size(16)));   // matches builtin param

// ---------------- CDNA5 async global->LDS helpers ----------------
__device__ __forceinline__ void cdna5_async_load_b128(const void* gsrc, void* ldst) {
#if defined(__HIP_DEVICE_COMPILE__)
#if __has_builtin(__builtin_amdgcn_global_load_async_to_lds_b128)
  __builtin_amdgcn_global_load_async_to_lds_b128(
      (__attribute__((address_space(1))) v4i_vs*)(unsigned long long)gsrc,
      (__attribute__((address_space(3))) v4i_vs*)(unsigned)(unsigned long long)ldst,
      0, 0);
#else
  // VGLOBAL async form: VDST = LDS byte address, VADDR = 64-bit global address
  asm volatile("global_load_async_to_lds_b128 %0, %1, off"
               :
               : "v"((unsigned)(unsigned long long)ldst),
                 "v"((unsigned long long)gsrc)
               : "memory");
#endif
#endif
}

// Wait until at most N async ops remain outstanding (threshold wait on ASYNCcnt)
#if defined(__HIP_DEVICE_COMPILE__)
#if __has_builtin(__builtin_amdgcn_s_wait_asynccnt)
#define CDNA5_WAIT_ASYNC(n)                       \
  do {                                            \
    __builtin_amdgcn_s_wait_asynccnt(n);          \
    asm volatile("" ::: "memory");                \
  } while (0)
#else
#define CDNA5_WAIT_ASYNC(n)                              \
  do {                                                   \
    asm volatile("s_wait_asynccnt %0" ::"i"(n) : "memory"); \
  } while (0)
#endif
#else
#define CDNA5_WAIT_ASYNC(n) do {} while (0)
#endif

// ---------------- Phase 1: ruler = max over 160 logits -> sortable key ------
// One wave handles 4 rows (2560 B) per tile. Double-buffered async pipeline:
// issue 5 wave-wide async b128 copies for tile t+1, s_wait_asynccnt 5 (tile t
// complete), reduce tile t from LDS with ds_load_b128 + wave32 shfl_xor.
__global__ __launch_bounds__(256) void ruler_keys_kernel(
    const float* __restrict__ cls, unsigned* __restrict__ keys, int nrows) {
  __shared__ __align__(16) float tile[8 * 2 * 640];  // 8 waves * 2 bufs * 640
  const int lane = threadIdx.x & 31;
  const int wib  = threadIdx.x >> 5;
  float* lds0 = tile + wib * 1280;
  const int waveId = (int)((blockIdx.x * blockDim.x + threadIdx.x) >> 5);
  const int nwaves = (int)((gridDim.x * blockDim.x) >> 5);
  const int ntiles = nrows >> 2;                  // all level sizes % 4 == 0

  int buf = 0;
  int t = waveId;
  if (t < ntiles) {
    const float* src = cls + (size_t)t * 640;
#pragma unroll
    for (int i = 0; i < 5; ++i)                   // 5 * 32 lanes * 16B = 2560B
      cdna5_async_load_b128(src + i * 128 + lane * 4, lds0 + i * 128 + lane * 4);
  }
  for (; t < ntiles; t += nwaves) {
    const int tn = t + nwaves;
    if (tn < ntiles) {
      const float* srcn = cls + (size_t)tn * 640;
      float* ldsn = lds0 + (buf ^ 1) * 640;
#pragma unroll
      for (int i = 0; i < 5; ++i)
        cdna5_async_load_b128(srcn + i * 128 + lane * 4, ldsn + i * 128 + lane * 4);
      CDNA5_WAIT_ASYNC(5);   // newest 5 may fly; tile t's 5 are complete
    } else {
      CDNA5_WAIT_ASYNC(0);
    }
    const float* lds = lds0 + buf * 640;
    const int row = lane >> 3;                    // 4 rows, 8 lanes each
    const int sub = lane & 7;
    const float4* q = (const float4*)(lds + row * 160 + sub * 20);
    float m = -3.4e38f;
#pragma unroll
    for (int i = 0; i < 5; ++i) {
      float4 v = q[i];
      m = fmaxf(m, fmaxf(fmaxf(v.x, v.y), fmaxf(v.z, v.w)));
    }
    m = fmaxf(m, __shfl_xor(m, 1));
    m = fmaxf(m, __shfl_xor(m, 2));
    m = fmaxf(m, __shfl_xor(m, 4));
    if (sub == 0) {
      unsigned b = __float_as_uint(m);
      keys[t * 4 + row] = (b & 0x80000000u) ? ~b : (b | 0x80000000u);
    }
    buf ^= 1;
  }
}

// ---------------- Phase 2: radix select (threshold of 1000th largest) -------
__global__ __launch_bounds__(256) void init_kernel(unsigned* __restrict__ hist,
                                                   unsigned* __restrict__ state,
                                                   unsigned* __restrict__ itemsU) {
  const int gid = blockIdx.x * blockDim.x + threadIdx.x;
  const int nth = gridDim.x * blockDim.x;
  for (int i = gid; i < 1024; i += nth) hist[i] = 0;
  for (int i = gid; i < 8192; i += nth) itemsU[i] = 0;   // 4 levels * 1024 u64
  if (gid < 4) {
    unsigned* s = state + gid * 8;
    s[0] = 0;        // prefix
    s[1] = TOPK_N;   // needed
    s[2] = 0;        // numGreater
    s[3] = 0;        // cntG
    s[4] = 0;        // cntE
  }
}

__global__ __launch_bounds__(256) void hist_kernel(const unsigned* __restrict__ keys,
                                                   int n,
                                                   const unsigned* __restrict__ state,
                                                   unsigned* __restrict__ hist,
                                                   int shift) {
  __shared__ unsigned lh[256];
  lh[threadIdx.x] = 0;
  __syncthreads();
  const unsigned prefix = state[0];
  const unsigned mask = (shift == 24) ? 0u : (0xFFFFFFFFu << (shift + 8));
  const int nth = (int)(gridDim.x * blockDim.x);
  for (int i = (int)(blockIdx.x * blockDim.x + threadIdx.x); i < n; i += nth) {
    unsigned k = keys[i];
    if ((k & mask) == prefix) atomicAdd(&lh[(k >> shift) & 255u], 1u);
  }
  __syncthreads();
  unsigned h = lh[threadIdx.x];
  if (h) atomicAdd(&hist[threadIdx.x], h);
}

__global__ __launch_bounds__(64) void scan_kernel(unsigned* __restrict__ state4,
                                                  unsigned* __restrict__ hist4,
                                                  int shift) {
  unsigned* state = state4 + blockIdx.x * 8;
  unsigned* hist  = hist4 + blockIdx.x * 256;
  if (threadIdx.x == 0) {
    unsigned needed = state[1], cum = 0;
    int b = 255;
    for (; b > 0; --b) {
      unsigned h = hist[b];
      if (cum + h >= needed) break;
      cum += h;
    }
    state[0] |= ((unsigned)b) << shift;
    state[1] = needed - cum;
    state[2] += cum;
  }
  __syncthreads();
  for (int i = threadIdx.x; i < 256; i += blockDim.x) hist[i] = 0;  // next pass
}

// ---------------- Phase 3: gather winners --------------------------------
__global__ __launch_bounds__(256) void gather_kernel(const unsigned* __restrict__ keys,
                                                     int n,
                                                     unsigned* __restrict__ state,
                                                     unsigned long long* __restrict__ items) {
  const unsigned T    = state[0];
  const unsigned ties = state[1];   // #(== T) to take
  const unsigned g    = state[2];   // exact #(> T), <= 999
  const int nth = (int)(gridDim.x * blockDim.x);
  for (int i = (int)(blockIdx.x * blockDim.x + threadIdx.x); i < n; i += nth) {
    unsigned k = keys[i];
    if (k > T) {
      unsigned p = atomicAdd(&state[3], 1u);
      if (p < 1024u)
        items[p] = ((unsigned long long)k << 32) | (unsigned)(~(unsigned)i);
    } else if (k == T) {
      unsigned p = atomicAdd(&state[4], 1u);
      if (p < ties && (g + p) < 1024u)
        items[g + p] = ((unsigned long long)k << 32) | (unsigned)(~(unsigned)i);
    }
  }
}

// ---------------- Phase 4: sort 1024 candidates, emit indices -------------
__global__ __launch_bounds__(1024) void sort_kernel(const unsigned long long* __restrict__ items4,
                                                    int* __restrict__ sel) {
  __shared__ unsigned long long s[1024];
  const int tid = threadIdx.x;
  const int l = blockIdx.x;
  s[tid] = items4[l * 1024 + tid];
  for (unsigned k = 2; k <= 1024u; k <<= 1) {
    for (unsigned j = k >> 1; j > 0; j >>= 1) {
      __syncthreads();
      unsigned ixj = (unsigned)tid ^ j;
      if (ixj > (unsigned)tid) {
        unsigned long long x = s[tid], y = s[ixj];
        bool desc = ((tid & k) == 0);
        if (desc ? (x < y) : (x > y)) { s[tid] = y; s[ixj] = x; }
      }
    }
  }
  __syncthreads();
  if (tid < TOPK_N) sel[l * TOPK_N + tid] = (int)(~(unsigned)s[tid]);
}

// ---------------- Phase 5: decode + emit [N*80, 12] -----------------------
__global__ __launch_bounds__(256) void finalize_kernel(
    const float* __restrict__ a0, const float* __restrict__ a1,
    const float* __restrict__ a2, const float* __restrict__ a3,
    const float* __restrict__ a4,
    const float* __restrict__ c0, const float* __restrict__ c1,
    const float* __restrict__ c2, const float* __restrict__ c3,
    const float* __restrict__ c4,
    const float* __restrict__ r0, const float* __restrict__ r1,
    const float* __restrict__ r2, const float* __restrict__ r3,
    const float* __restrict__ r4,
    const int* __restrict__ sel, float* __restrict__ out) {
  const int gid = (int)(blockIdx.x * blockDim.x + threadIdx.x);
  const int total = N_KEPT * CLASS_NUM;
  if (gid >= total) return;
  const int nrow = gid / CLASS_NUM;
  const int cls  = gid - nrow * CLASS_NUM;

  int level, src;
  if (nrow < 4000) { level = nrow / 1000; src = sel[nrow]; }
  else             { level = 4;           src = nrow - 4000; }

  const float *A, *Cc, *R;
  switch (level) {
    case 0: A = a0; Cc = c0; R = r0; break;
    case 1: A = a1; Cc = c1; R = r1; break;
    case 2: A = a2; Cc = c2; R = r2; break;
    case 3: A = a3; Cc = c3; R = r3; break;
    default: A = a4; Cc = c4; R = r4; break;
  }
  const float4 av  = ((const float4*)A)[src];
  const float4 rv0 = ((const float4*)R)[(size_t)src * 2];
  const float4 rv1 = ((const float4*)R)[(size_t)src * 2 + 1];
  const float sc0 = Cc[(size_t)src * 160 + cls];
  const float sc1 = Cc[(size_t)src * 160 + 80 + cls];

  const float w  = av.z - av.x + 1.f;
  const float h  = av.w - av.y + 1.f;
  const float cx = av.x + 0.5f * w;
  const float cy = av.y + 0.5f * h;
  const float tag = (float)(cls + 1);
  const float s0 = 1.f / (1.f + expf(-sc0));
  const float s1 = 1.f / (1.f + expf(-sc1));

  const float pcx0 = cx + rv0.x * w, pcy0 = cy + rv0.y * h;
  const float pw0 = w * expf(rv0.z), ph0 = h * expf(rv0.w);
  const float pcx1 = cx + rv1.x * w, pcy1 = cy + rv1.y * h;
  const float pw1 = w * expf(rv1.z), ph1 = h * expf(rv1.w);

  float4* op = (float4*)(out + (size_t)gid * 12);   // 48B row -> 16B aligned
  op[0] = make_float4(pcx0 - 0.5f * pw0, pcy0 - 0.5f * ph0,
                      pcx0 + 0.5f * pw0, pcy0 + 0.5f * ph0);
  op[1] = make_float4(s0, tag, pcx1 - 0.5f * pw1, pcy1 - 0.5f * ph1);
  op[2] = make_float4(pcx1 + 0.5f * pw1, pcy1 + 0.5f * ph1, s1, tag);
}

// ---------------- host ----------------------------------------------------
extern "C" void kernel_launch(void* const* d_in, const int* in_sizes, int n_in,
                              void* d_out, int out_size, void* d_ws, size_t ws_size,
                              hipStream_t stream) {
  (void)in_sizes; (void)n_in; (void)out_size; (void)ws_size;
  // setup_inputs() dict order: anchors0, cls0, reg0, anchors1, cls1, reg1, ...
  const float* A[5]; const float* C[5]; const float* R[5];
  for (int i = 0; i < 5; ++i) {
    A[i] = (const float*)d_in[3 * i];
    C[i] = (const float*)d_in[3 * i + 1];
    R[i] = (const float*)d_in[3 * i + 2];
  }
  // Workspace layout (uint words): keys[195840] | hist[1024] | state[32] | sel[4000] | items[4*1024 u64]
  unsigned* keys  = (unsigned*)d_ws;
  unsigned* hist  = keys + KEYS_TOTAL;
  unsigned* state = hist + 1024;
  int*      sel   = (int*)(state + 32);
  unsigned long long* items =
      (unsigned long long*)((char*)d_ws + (size_t)(KEYS_TOTAL + 1024 + 32 + 4000) * 4);

  init_kernel<<<16, 256, 0, stream>>>(hist, state, (unsigned*)items);

  for (int l = 0; l < 4; ++l) {
    int ntiles = LVL_N[l] >> 2;
    int blocks = (ntiles + 7) / 8;
    if (blocks > 4096) blocks = 4096;
    ruler_keys_kernel<<<blocks, 256, 0, stream>>>(C[l], keys + KOFF[l], LVL_N[l]);
  }

  const int shifts[4] = {24, 16, 8, 0};
  for (int p = 0; p < 4; ++p) {
    for (int l = 0; l < 4; ++l) {
      int hb = (LVL_N[l] + 255) / 256;
      if (hb > 1024) hb = 1024;
      hist_kernel<<<hb, 256, 0, stream>>>(keys + KOFF[l], LVL_N[l],
                                          state + 8 * l, hist + 256 * l, shifts[p]);
    }
    scan_kernel<<<4, 64, 0, stream>>>(state, hist, shifts[p]);
  }

  for (int l = 0; l < 4; ++l) {
    int gb = (LVL_N[l] + 255) / 256;
    if (gb > 1024) gb = 1024;
    gather_kernel<<<gb, 256, 0, stream>>>(keys + KOFF[l], LVL_N[l],
                                          state + 8 * l, items + 1024 * l);
  }

  sort_kernel<<<4, 1024, 0, stream>>>(items, sel);

  const int total = N_KEPT * CLASS_NUM;
  finalize_kernel<<<(total + 255) / 256, 256, 0, stream>>>(
      A[0], A[1], A[2], A[3], A[4],
      C[0], C[1], C[2], C[3], C[4],
      R[0], R[1], R[2], R[3], R[4],
      sel, (float*)d_out);
}